// EMAVectorQuantizer_65841848648439
// MI455X (gfx1250) — compile-verified
//
#include <hip/hip_runtime.h>

typedef __attribute__((ext_vector_type(2))) float v2f;
typedef __attribute__((ext_vector_type(8))) float v8f;

#define C_DIM   256
#define K_CODES 8192
#define N_VECS  32768
#define HW      1024          // H*W
#define Z_PER_B (C_DIM * HW)  // 262144 floats per image

// ---- output layout (floats, concatenated in reference return order) ----
#define QOFF     0            // quantized_st [B,C,H,W]  (8388608)
#define IDXOFF   8388608      // indices [N]             (32768, written as float)
#define LOSSOFF  8421376      // vq_loss                 (1)
#define PERPOFF  8421377      // perplexity              (1)
#define ACTOFF   8421378      // active_codes            (1)
#define USAGEOFF 8421379      // usage_counts [K]        (8192)
#define ESUMOFF  8429571      // embed_sum [K,C]         (2097152)
#define Q_ELEMS  8388608

// ---- workspace layout (float offsets into d_ws) ----
#define WS_CBNORM 0
#define WS_IDX    8192                // int[N]
#define WS_LOSS   (8192 + 32768)     // float

#define VSTRIDE 260   // LDS row stride for staged vectors (conflict-free frag loads)

// ---------------------------------------------------------------- zero tail
__global__ void vq_zero(float* __restrict__ out_tail, float* __restrict__ ws_loss, int n) {
  int i = blockIdx.x * blockDim.x + threadIdx.x;
  if (i < n) out_tail[i] = 0.0f;
  if (i == 0) *ws_loss = 0.0f;
}

// ----------------------------------------------------- codebook row norms
__global__ __launch_bounds__(256) void vq_cbnorm(const float* __restrict__ cb,
                                                 float* __restrict__ cbnorm) {
  int wave = threadIdx.x >> 5;
  int lane = threadIdx.x & 31;
  int row  = blockIdx.x * 8 + wave;
  const float* p = cb + (size_t)row * C_DIM;
  float s = 0.0f;
  for (int c = lane; c < C_DIM; c += 32) { float v = p[c]; s += v * v; }
  for (int m = 16; m >= 1; m >>= 1) s += __shfl_xor(s, m, 32);
  if (lane == 0) cbnorm[row] = s;
}

// ------------------------------------- fused distance GEMM + argmin (WMMA)
// A = codebook tiles (16x4 f32), B = staged vector tiles (4x16 f32).
// Register tile: 4 code tiles x 2 vector tiles = 8 accumulators per wave,
// so each k-step issues 4 global b64 + 2 LDS b64 for 8 v_wmma_f32_16x16x4_f32.
__global__ __launch_bounds__(256) void vq_argmin(const float* __restrict__ z,
                                                 const float* __restrict__ cb,
                                                 const float* __restrict__ cbnorm,
                                                 int*   __restrict__ wsIdx,
                                                 float* __restrict__ outIdx,
                                                 float* __restrict__ usage) {
  __shared__ float Vlds[32 * VSTRIDE];
  __shared__ float RedV[32][16];
  __shared__ int   RedI[32][16];

  const int tid  = threadIdx.x;
  const int nb   = blockIdx.x * 32;      // first vector handled by this block
  const int bimg = nb >> 10;             // image index (nb aligned: same image)
  const int rem  = nb & 1023;            // position within H*W
  const size_t zbase = (size_t)bimg * Z_PER_B + rem;

  // stage 32 vectors x 256 dims: z[b,c,hw] is coalesced over dn for fixed c
  #pragma unroll
  for (int i = 0; i < 32; ++i) {
    int f = tid + i * 256;
    int c = f >> 5, dn = f & 31;
    Vlds[dn * VSTRIDE + c] = z[zbase + (size_t)c * HW + dn];
  }
  __syncthreads();

  const int wave  = tid >> 5;
  const int lane  = tid & 31;
  const int half  = lane >> 4;   // k-split half of the wave
  const int col16 = lane & 15;

  float minv0 = 3.4e38f, minv1 = 3.4e38f;
  int   mini0 = 0,       mini1 = 0;

  const float* b0p = &Vlds[col16 * VSTRIDE + half * 2];
  const float* b1p = &Vlds[(16 + col16) * VSTRIDE + half * 2];

  // 512 code tiles total; each wave owns groups of 4 consecutive tiles.
  for (int g = wave * 4; g < K_CODES / 16; g += 32) {
    const int cb0 = g * 16;  // first code of this 64-code group
    v8f acc0[4] = {{}, {}, {}, {}};   // D[code][vec tile 0]
    v8f acc1[4] = {{}, {}, {}, {}};   // D[code][vec tile 1]

    const float* ap[4];
    #pragma unroll
    for (int at = 0; at < 4; ++at)
      ap[at] = cb + (size_t)(cb0 + at * 16 + col16) * C_DIM + half * 2;

    // prefetch next group's codebook rows into near caches
    if (g + 32 < K_CODES / 16) {
      #pragma unroll
      for (int at = 0; at < 4; ++at)
        __builtin_prefetch(ap[at] + (size_t)32 * 16 * C_DIM, 0, 1);
    }

    #pragma unroll 4
    for (int kk = 0; kk < 64; ++kk) {
      v2f b0 = *(const v2f*)(b0p + kk * 4);
      v2f b1 = *(const v2f*)(b1p + kk * 4);
      #pragma unroll
      for (int at = 0; at < 4; ++at) {
        v2f a = *(const v2f*)(ap[at] + kk * 4);
        acc0[at] = __builtin_amdgcn_wmma_f32_16x16x4_f32(false, a, false, b0,
                                                         (short)0, acc0[at], false, false);
        acc1[at] = __builtin_amdgcn_wmma_f32_16x16x4_f32(false, a, false, b1,
                                                         (short)0, acc1[at], false, false);
      }
    }

    // score = ||c||^2 - 2*dot ; D-layout: VGPR r => code row r + 8*half.
    // Scan order is ascending in code index -> strict '<' keeps lowest index.
    #pragma unroll
    for (int at = 0; at < 4; ++at) {
      const int base = cb0 + at * 16 + half * 8;
      const float* cnp = cbnorm + base;
      #pragma unroll
      for (int r = 0; r < 8; ++r) {
        float cn   = cnp[r];
        int   code = base + r;
        float s0 = cn - 2.0f * acc0[at][r];
        float s1 = cn - 2.0f * acc1[at][r];
        if (s0 < minv0) { minv0 = s0; mini0 = code; }
        if (s1 < minv1) { minv1 = s1; mini1 = code; }
      }
    }
  }

  // cross-half / cross-wave argmin reduction (tie-break: lowest index)
  int slot = wave * 2 + half;
  RedV[col16][slot]      = minv0; RedI[col16][slot]      = mini0;
  RedV[16 + col16][slot] = minv1; RedI[16 + col16][slot] = mini1;
  __syncthreads();
  if (tid < 32) {
    float bv = RedV[tid][0]; int bi = RedI[tid][0];
    #pragma unroll
    for (int s = 1; s < 16; ++s) {
      float v = RedV[tid][s]; int ix = RedI[tid][s];
      if (v < bv || (v == bv && ix < bi)) { bv = v; bi = ix; }
    }
    int n = nb + tid;
    wsIdx[n]  = bi;
    outIdx[n] = (float)bi;
    atomicAdd(usage + bi, 1.0f);
  }
}

// --------------------- gather quantized, vq_loss partials, embed_sum atomics
__global__ __launch_bounds__(256) void vq_gather(const float* __restrict__ z,
                                                 const float* __restrict__ cb,
                                                 const int*   __restrict__ wsIdx,
                                                 float* __restrict__ outQ,
                                                 float* __restrict__ esum,
                                                 float* __restrict__ lossAcc) {
  __shared__ int   idxs[64];
  __shared__ float red[256];
  const int tid = threadIdx.x;
  const int nb  = blockIdx.x * 64;
  if (tid < 64) idxs[tid] = wsIdx[nb + tid];
  __syncthreads();
  const int bimg = nb >> 10, rem = nb & 1023;
  const size_t zbase = (size_t)bimg * Z_PER_B + rem;
  float acc = 0.0f;
  for (int i = 0; i < 64; ++i) {
    int f = tid + i * 256;
    int c = f >> 6, dn = f & 63;
    size_t zi = zbase + (size_t)c * HW + dn;
    float zv = z[zi];
    int   idx = idxs[dn];
    float q = cb[(size_t)idx * C_DIM + c];
    outQ[zi] = q;                    // quantized_st == quantized numerically
    float d = q - zv;
    acc += d * d;
    atomicAdd(esum + (size_t)idx * C_DIM + c, zv);
  }
  red[tid] = acc;
  __syncthreads();
  for (int s = 128; s > 0; s >>= 1) {
    if (tid < s) red[tid] += red[tid + s];
    __syncthreads();
  }
  if (tid == 0) atomicAdd(lossAcc, red[0]);
}

// ------------------------------------------- perplexity / active / mean loss
__global__ __launch_bounds__(256) void vq_finalize(const float* __restrict__ usage,
                                                   const float* __restrict__ lossAcc,
                                                   float* __restrict__ out) {
  __shared__ float redT[256];
  __shared__ float redP[256];
  __shared__ float redA[256];
  int tid = threadIdx.x;
  float t = 0.0f;
  for (int k = tid; k < K_CODES; k += 256) t += usage[k];
  redT[tid] = t; __syncthreads();
  for (int s = 128; s > 0; s >>= 1) { if (tid < s) redT[tid] += redT[tid + s]; __syncthreads(); }
  float denom = fmaxf(redT[0], 1.0f);
  float p = 0.0f, a = 0.0f;
  for (int k = tid; k < K_CODES; k += 256) {
    float u  = usage[k];
    float pr = u / denom;
    p += pr * logf(fmaxf(pr, 1e-10f));
    a += (u > 0.0f) ? 1.0f : 0.0f;
  }
  redP[tid] = p; redA[tid] = a; __syncthreads();
  for (int s = 128; s > 0; s >>= 1) {
    if (tid < s) { redP[tid] += redP[tid + s]; redA[tid] += redA[tid + s]; }
    __syncthreads();
  }
  if (tid == 0) {
    out[LOSSOFF] = lossAcc[0] / (float)Q_ELEMS;
    out[PERPOFF] = expf(-redP[0]);
    out[ACTOFF]  = redA[0];
  }
}

extern "C" void kernel_launch(void* const* d_in, const int* in_sizes, int n_in,
                              void* d_out, int out_size, void* d_ws, size_t ws_size,
                              hipStream_t stream) {
  const float* z  = (const float*)d_in[0];   // [32,256,32,32]
  const float* cb = (const float*)d_in[1];   // [8192,256]
  float* out = (float*)d_out;
  float* ws  = (float*)d_ws;

  float* cbnorm  = ws + WS_CBNORM;
  int*   wsIdx   = (int*)(ws + WS_IDX);
  float* lossAcc = ws + WS_LOSS;

  const int ztail = 3 + K_CODES + K_CODES * C_DIM;  // loss..embed_sum region
  vq_zero<<<(ztail + 255) / 256, 256, 0, stream>>>(out + LOSSOFF, lossAcc, ztail);
  vq_cbnorm<<<K_CODES / 8, 256, 0, stream>>>(cb, cbnorm);
  vq_argmin<<<N_VECS / 32, 256, 0, stream>>>(z, cb, cbnorm, wsIdx,
                                             out + IDXOFF, out + USAGEOFF);
  vq_gather<<<N_VECS / 64, 256, 0, stream>>>(z, cb, wsIdx, out + QOFF,
                                             out + ESUMOFF, lossAcc);
  vq_finalize<<<1, 256, 0, stream>>>(out + USAGEOFF, lossAcc, out);
}